// ResnetBlockFC_Hyper_72962904424979
// MI455X (gfx1250) — compile-verified
//
#include <hip/hip_runtime.h>

#define B_SCENES 64
#define N_PTS    16384
#define SIZE_IN  256
#define SIZE_H   128
#define SIZE_OUT 128

typedef float v2f __attribute__((ext_vector_type(2)));
typedef float v8f __attribute__((ext_vector_type(8)));

// ---------------- prep kernels: bucket points by scene ----------------

__global__ void k_zero(int* __restrict__ count) {
  if (threadIdx.x < B_SCENES) count[threadIdx.x] = 0;
}

__global__ void k_hist(const long long* __restrict__ bi, int* __restrict__ count, int n) {
  int i = blockIdx.x * blockDim.x + threadIdx.x;
  if (i < n) atomicAdd(&count[(int)bi[i]], 1);
}

__global__ void k_scan(const int* __restrict__ count,
                       int* __restrict__ offsets, int* __restrict__ cursor) {
  if (threadIdx.x == 0) {
    int acc = 0;
    for (int i = 0; i < B_SCENES; ++i) {
      offsets[i] = acc; cursor[i] = acc; acc += count[i];
    }
    offsets[B_SCENES] = acc;
  }
}

__global__ void k_perm(const long long* __restrict__ bi, int* __restrict__ cursor,
                       int* __restrict__ perm, int n) {
  int i = blockIdx.x * blockDim.x + threadIdx.x;
  if (i < n) {
    int pos = atomicAdd(&cursor[(int)bi[i]], 1);
    perm[pos] = i;
  }
}

// ---------------- main compute: WMMA fp32 residual MLP ----------------
// grid.x = scene, grid.y = tile chunk; 128 threads = 4 waves, 1 tile of
// 16 points per wave iteration. Each wave owns an LDS slab for the
// D-layout -> A-layout transpose of the hidden activations.

__launch_bounds__(128)
__global__ void k_resblock(const float* __restrict__ W0,   // [B][128][256]
                           const float* __restrict__ W1,   // [B][128][128]
                           const float* __restrict__ W2,   // [B][128][256]
                           const float* __restrict__ B0,   // [B][128]
                           const float* __restrict__ B1,   // [B][128]
                           const float* __restrict__ X,    // [N][256]
                           const int* __restrict__ offsets,// [B+1]
                           const int* __restrict__ perm,   // [N]
                           float* __restrict__ OUT)        // [N][128]
{
  constexpr int HS = 132;              // padded LDS row stride (floats): conflict-free
  __shared__ float Hs[4][16 * HS];     // 33792 bytes, one slab per wave

  const int b   = blockIdx.x;
  const int beg = offsets[b];
  const int cnt = offsets[b + 1] - beg;
  if (cnt <= 0) return;

  const int lane = threadIdx.x & 31;
  const int wave = threadIdx.x >> 5;
  const int half = lane >> 4;          // 0: lanes 0-15, 1: lanes 16-31
  const int lcol = lane & 15;

  float* Hw = Hs[wave];

  const float* w0 = W0 + (size_t)b * SIZE_H   * SIZE_IN;
  const float* w1 = W1 + (size_t)b * SIZE_OUT * SIZE_H;
  const float* w2 = W2 + (size_t)b * SIZE_OUT * SIZE_IN;
  const float* bias0 = B0 + (size_t)b * SIZE_H;
  const float* bias1 = B1 + (size_t)b * SIZE_OUT;

  const int numTiles = (cnt + 15) >> 4;
  for (int t = blockIdx.y * 4 + wave; t < numTiles; t += gridDim.y * 4) {
    // A-fragment row for this lane (clamped for the ragged last tile; WMMA
    // needs EXEC all-ones so we compute garbage rows and just don't store them)
    int gmA = t * 16 + lcol;
    if (gmA > cnt - 1) gmA = cnt - 1;
    const int rowA = perm[beg + gmA];
    const float* xrow = X + (size_t)rowA * SIZE_IN;

    // ---- layer 0: H(16x128) = relu(X) * W0^T, 8 WMMA column tiles
    v8f accH[8] = {};
    for (int k = 0; k < SIZE_IN; k += 4) {
      v2f a = *(const v2f*)(xrow + k + 2 * half);
      a.x = fmaxf(a.x, 0.f);
      a.y = fmaxf(a.y, 0.f);
      #pragma unroll
      for (int nt = 0; nt < 8; ++nt) {
        const int h = nt * 16 + lcol;
        v2f bf = *(const v2f*)(w0 + (size_t)h * SIZE_IN + k + 2 * half);
        accH[nt] = __builtin_amdgcn_wmma_f32_16x16x4_f32(
            false, a, false, bf, (short)0, accH[nt], false, false);
      }
    }

    // bias0 + relu, spill H to this wave's LDS slab ([m][h], stride HS)
    #pragma unroll
    for (int nt = 0; nt < 8; ++nt) {
      const int h = nt * 16 + lcol;
      const float bb = bias0[h];
      #pragma unroll
      for (int r = 0; r < 8; ++r) {
        const int m = r + 8 * half;
        Hw[m * HS + h] = fmaxf(accH[nt][r] + bb, 0.f);
      }
    }

    // ---- layer 1 (K=128, A from LDS) + shortcut (K=256, A = raw x) fused
    v8f acc[8] = {};
    for (int k = 0; k < SIZE_H; k += 4) {
      v2f a = *(const v2f*)(&Hw[lcol * HS + k + 2 * half]);
      #pragma unroll
      for (int nt = 0; nt < 8; ++nt) {
        const int o = nt * 16 + lcol;
        v2f bf = *(const v2f*)(w1 + (size_t)o * SIZE_H + k + 2 * half);
        acc[nt] = __builtin_amdgcn_wmma_f32_16x16x4_f32(
            false, a, false, bf, (short)0, acc[nt], false, false);
      }
    }
    for (int k = 0; k < SIZE_IN; k += 4) {
      v2f a = *(const v2f*)(xrow + k + 2 * half);   // no relu on shortcut
      #pragma unroll
      for (int nt = 0; nt < 8; ++nt) {
        const int o = nt * 16 + lcol;
        v2f bf = *(const v2f*)(w2 + (size_t)o * SIZE_IN + k + 2 * half);
        acc[nt] = __builtin_amdgcn_wmma_f32_16x16x4_f32(
            false, a, false, bf, (short)0, acc[nt], false, false);
      }
    }

    // bias1 + scatter back to the original point ordering
    #pragma unroll
    for (int nt = 0; nt < 8; ++nt) {
      const int o = nt * 16 + lcol;
      const float bb = bias1[o];
      #pragma unroll
      for (int r = 0; r < 8; ++r) {
        const int m  = r + 8 * half;
        const int gm = t * 16 + m;
        if (gm < cnt) {
          const int row = perm[beg + gm];
          OUT[(size_t)row * SIZE_OUT + o] = acc[nt][r] + bb;
        }
      }
    }
  }
}

// ---------------- launcher ----------------

extern "C" void kernel_launch(void* const* d_in, const int* in_sizes, int n_in,
                              void* d_out, int out_size, void* d_ws, size_t ws_size,
                              hipStream_t stream) {
  (void)in_sizes; (void)n_in; (void)out_size; (void)ws_size;
  const float*     kls0  = (const float*)d_in[0];
  const float*     kls1  = (const float*)d_in[1];
  const float*     kls2  = (const float*)d_in[2];
  const float*     bias0 = (const float*)d_in[3];
  const float*     bias1 = (const float*)d_in[4];
  const float*     x     = (const float*)d_in[5];
  const long long* bi    = (const long long*)d_in[6];
  float* out = (float*)d_out;

  int* count   = (int*)d_ws;               // [64]
  int* offsets = count + B_SCENES;         // [65]
  int* cursor  = offsets + (B_SCENES + 1); // [64]
  int* perm    = cursor + B_SCENES;        // [16384]

  k_zero<<<1, 64, 0, stream>>>(count);
  k_hist<<<(N_PTS + 255) / 256, 256, 0, stream>>>(bi, count, N_PTS);
  k_scan<<<1, 1, 0, stream>>>(count, offsets, cursor);
  k_perm<<<(N_PTS + 255) / 256, 256, 0, stream>>>(bi, cursor, perm, N_PTS);

  dim3 grid(B_SCENES, 8);
  k_resblock<<<grid, 128, 0, stream>>>(kls0, kls1, kls2, bias0, bias1, x,
                                       offsets, perm, out);
}